// TRSeparateWeightTensorProduct_71356586656070
// MI455X (gfx1250) — compile-verified
//
#include <hip/hip_runtime.h>

typedef __attribute__((ext_vector_type(2))) float v2f;
typedef __attribute__((ext_vector_type(8))) float v8f;

#define MUL 128
#define MO  128
#define XW  (MUL + 3 * MUL) /* 512 floats per x row / out row */
#define NSLICE 8            /* 128 cols / 16 per wave */
#define WG_PER_SLICE 32
#define WAVES_PER_WG 8

__device__ __forceinline__ v8f wmma4(v2f a, v2f b, v8f c) {
  // 8 args: (neg_a, A, neg_b, B, c_mod, C, reuse_a, reuse_b)
  return __builtin_amdgcn_wmma_f32_16x16x4_f32(
      false, a, false, b, (short)0, c, false, false);
}

struct F3 { float x, y, z; };

__global__ __launch_bounds__(256)
void tp_wmma_kernel(const float* __restrict__ x1, const float* __restrict__ x2,
                    const float* __restrict__ w1_000, const float* __restrict__ w2_000,
                    const float* __restrict__ w1_110, const float* __restrict__ w2_110,
                    const float* __restrict__ w1_011, const float* __restrict__ w2_011,
                    const float* __restrict__ w1_101, const float* __restrict__ w2_101,
                    float* __restrict__ out, int Bsz, int tiles)
{
  // Transposed weight slices: [weight][col(16)][k(128)+pad] ; pad=2 -> conflict-free,
  // (c*130 + even k)*4 is 8-byte aligned for ds_load_b64 fragments.
  __shared__ float lds[8][16][130];

  const float* wp[8] = {w1_000, w2_000, w1_110, w2_110,
                        w1_011, w2_011, w1_101, w2_101};

  const int n     = blockIdx.x % NSLICE;
  const int g     = blockIdx.x / NSLICE;
  const int tid   = threadIdx.x;
  const int lane  = tid & 31;
  const int wave  = tid >> 5;
  const int ncol0 = n * 16;

  // Cooperative one-time weight slice load (64 KB/WG), stored transposed.
  for (int idx = tid; idx < 8 * 128 * 16; idx += 256) {
    int c  = idx & 15;
    int k  = (idx >> 4) & 127;
    int wi = idx >> 11;
    lds[wi][c][k] = wp[wi][k * MO + ncol0 + c];
  }
  __syncthreads();

  const int c     = lane & 15;  // N within slice (also M for A frags)
  const int khalf = lane >> 4;  // lane half selects K pair (per f32 WMMA layout)

  const float c0 = 5.5242717280199024e-03f; // A0  (== A1 * INV_SQRT3)
  const float c1 = 3.1894420157944281e-03f; // A0 * INV_SQRT3

  const int stride = WG_PER_SLICE * WAVES_PER_WG;

  for (int t = g * WAVES_PER_WG + wave; t < tiles; t += stride) {
    int rowA = t * 16 + c;
    if (rowA >= Bsz) rowA = Bsz - 1;            // tail-safe loads
    const float* x1r = x1 + (size_t)rowA * XW;
    const float* x2r = x2 + (size_t)rowA * XW;

    // Hint next tile's rows toward the caches.
    int tn = t + stride;
    if (tn < tiles) {
      int rn = tn * 16 + c;
      if (rn < Bsz) {
        __builtin_prefetch(x1 + (size_t)rn * XW, 0, 0);
        __builtin_prefetch(x2 + (size_t)rn * XW, 0, 0);
        __builtin_prefetch(x1 + (size_t)rn * XW + 256, 0, 0);
        __builtin_prefetch(x2 + (size_t)rn * XW + 256, 0, 0);
      }
    }

    // 16 GEMM accumulators, each a 16x16 f32 tile (8 VGPRs).
    // 0: x1_0@w1_000   1: x2_0@w2_000   2..4: x1_1_i@w1_110  5..7: x2_1_i@w2_110
    // 8: x1_0@w1_011   9..11: x2_1_j@w2_011  12..14: x1_1_j@w1_101  15: x2_0@w2_101
    v8f acc[16] = {};

    for (int k = 0; k < 32; ++k) {
      const int kb = k * 4 + khalf * 2;

      // A fragments (f32 16x4 layout: M = lane%16, K = kb..kb+1)
      v2f a10 = *(const v2f*)(x1r + kb);
      v2f a20 = *(const v2f*)(x2r + kb);
      const float* p1 = x1r + MUL + 3 * kb;     // x1_1[row][kb][i] = x1[row][128+3*kb+i]
      const float* p2 = x2r + MUL + 3 * kb;
      v2f a11[3], a21[3];
#pragma unroll
      for (int i = 0; i < 3; ++i) {
        a11[i] = (v2f){p1[i], p1[i + 3]};
        a21[i] = (v2f){p2[i], p2[i + 3]};
      }

      // B fragments (f32 4x16 layout: N = lane%16, K rows kb..kb+1) from LDS.
      v2f b;
      b = *(const v2f*)&lds[0][c][kb]; acc[0]  = wmma4(a10, b, acc[0]);
      b = *(const v2f*)&lds[1][c][kb]; acc[1]  = wmma4(a20, b, acc[1]);
      b = *(const v2f*)&lds[2][c][kb];
#pragma unroll
      for (int i = 0; i < 3; ++i) acc[2 + i]  = wmma4(a11[i], b, acc[2 + i]);
      b = *(const v2f*)&lds[3][c][kb];
#pragma unroll
      for (int i = 0; i < 3; ++i) acc[5 + i]  = wmma4(a21[i], b, acc[5 + i]);
      b = *(const v2f*)&lds[4][c][kb]; acc[8]  = wmma4(a10, b, acc[8]);
      b = *(const v2f*)&lds[5][c][kb];
#pragma unroll
      for (int j = 0; j < 3; ++j) acc[9 + j]  = wmma4(a21[j], b, acc[9 + j]);
      b = *(const v2f*)&lds[6][c][kb];
#pragma unroll
      for (int j = 0; j < 3; ++j) acc[12 + j] = wmma4(a11[j], b, acc[12 + j]);
      b = *(const v2f*)&lds[7][c][kb]; acc[15] = wmma4(a20, b, acc[15]);
    }

    // Elementwise combine (all in f32 VGPRs).
    v8f o0  = c0 * (acc[0] * acc[1])
            + c1 * (acc[2] * acc[5] + acc[3] * acc[6] + acc[4] * acc[7]);
    v8f o1a = c0 * (acc[8] * acc[9]  + acc[12] * acc[15]);
    v8f o1b = c0 * (acc[8] * acc[10] + acc[13] * acc[15]);
    v8f o1c = c0 * (acc[8] * acc[11] + acc[14] * acc[15]);

    // C/D layout: VGPR e -> M = e + 8*khalf, N = lane%16.
#pragma unroll
    for (int e = 0; e < 8; ++e) {
      int row = t * 16 + e + 8 * khalf;
      if (row < Bsz) {
        float* orow = out + (size_t)row * XW;
        orow[ncol0 + c] = o0[e];
        F3 v; v.x = o1a[e]; v.y = o1b[e]; v.z = o1c[e];
        *(F3*)(orow + MUL + 3 * (ncol0 + c)) = v;   // out1[row][w][0..2], w = ncol0+c
      }
    }
  }
}

extern "C" void kernel_launch(void* const* d_in, const int* in_sizes, int n_in,
                              void* d_out, int out_size, void* d_ws, size_t ws_size,
                              hipStream_t stream) {
  // setup_inputs order: x1, x2, w1_000, w2_000, w1_011, w2_011, w1_101, w2_101, w1_110, w2_110
  const float* x1     = (const float*)d_in[0];
  const float* x2     = (const float*)d_in[1];
  const float* w1_000 = (const float*)d_in[2];
  const float* w2_000 = (const float*)d_in[3];
  const float* w1_011 = (const float*)d_in[4];
  const float* w2_011 = (const float*)d_in[5];
  const float* w1_101 = (const float*)d_in[6];
  const float* w2_101 = (const float*)d_in[7];
  const float* w1_110 = (const float*)d_in[8];
  const float* w2_110 = (const float*)d_in[9];
  float* out = (float*)d_out;

  int Bsz   = in_sizes[0] / XW;
  int tiles = (Bsz + 15) / 16;

  dim3 grid(NSLICE * WG_PER_SLICE);
  dim3 block(256);
  hipLaunchKernelGGL(tp_wmma_kernel, grid, block, 0, stream,
                     x1, x2, w1_000, w2_000, w1_110, w2_110,
                     w1_011, w2_011, w1_101, w2_101, out, Bsz, tiles);
}